// GCN_HIGH_58437325029846
// MI455X (gfx1250) — compile-verified
//
#include <hip/hip_runtime.h>
#include <math.h>

typedef long long i64;
typedef __attribute__((ext_vector_type(2))) float v2f;
typedef __attribute__((ext_vector_type(8))) float v8f;

#define HDIM 128

// ---------------------------------------------------------------------------
// Fold the two conv branches into one weight/bias set per layer:
//   layer0: Wc = 0.95*W1 + 0.05*Wr1 ; layers 1,2: Wc = 0.95*W2[i] + 0.05*Wr
// ---------------------------------------------------------------------------
__global__ void k_fold(const float* __restrict__ W1, const float* __restrict__ Wr1,
                       const float* __restrict__ W2, const float* __restrict__ Wr,
                       const float* __restrict__ b1, const float* __restrict__ br1,
                       const float* __restrict__ b2, const float* __restrict__ br,
                       float* __restrict__ Wc, float* __restrict__ bc) {
    int idx = blockIdx.x * 256 + threadIdx.x;
    const int WSZ = HDIM * HDIM;
    if (idx < WSZ) {
        Wc[idx] = 0.95f * W1[idx] + 0.05f * Wr1[idx];
    } else if (idx < 3 * WSZ) {
        int l = idx / WSZ - 1;
        int r = idx % WSZ;
        Wc[idx] = 0.95f * W2[(size_t)l * WSZ + r] + 0.05f * Wr[r];
    } else {
        int t = idx - 3 * WSZ;
        if (t < HDIM) {
            bc[t] = 0.95f * b1[t] + 0.05f * br1[t];
        } else if (t < 3 * HDIM) {
            int l = t / HDIM - 1;
            int r = t % HDIM;
            bc[t] = 0.95f * b2[l * HDIM + r] + 0.05f * br[r];
        }
    }
}

// deg[dst] += 1 per edge
__global__ void k_deg(const i64* __restrict__ dst, float* __restrict__ deg, int E) {
    int e = blockIdx.x * 256 + threadIdx.x;
    if (e < E) atomicAdd(&deg[(int)dst[e]], 1.0f);
}

// dinv = (deg+1)^-0.5 ; selfn = 1/(deg+1)
__global__ void k_norms(const float* __restrict__ deg, float* __restrict__ dinv,
                        float* __restrict__ selfn, int N) {
    int n = blockIdx.x * 256 + threadIdx.x;
    if (n < N) {
        float d = deg[n] + 1.0f;
        dinv[n]  = rsqrtf(d);
        selfn[n] = 1.0f / d;
    }
}

// ---------------------------------------------------------------------------
// z[N x 128] = (relu?)(h) @ Wc[128 x 128]  via V_WMMA_F32_16X16X4_F32.
// One wave per 16-row tile, owning all 8 column tiles (A frag reused 8x).
// Wc cached in LDS in k-pair interleaved layout:
//   sW[(k>>1)*256 + 2*col + (k&1)] = W[k][col]
// so a B fragment (W[ka][col], W[ka+1][col]) is ONE aligned ds_load_b64
// directly into the even-aligned VGPR pair WMMA consumes (no shuffles).
// ---------------------------------------------------------------------------
__global__ __launch_bounds__(256) void k_gemm_wmma(
        const float* __restrict__ h, const float* __restrict__ Wm,
        float* __restrict__ z, int nTiles, int N, int relu_in) {
    __shared__ float sW[HDIM * HDIM];
    for (int i = threadIdx.x; i < HDIM * HDIM; i += 256) {
        int r = i >> 7;            // k row
        int c = i & (HDIM - 1);    // col
        sW[(r >> 1) * (2 * HDIM) + (c << 1) + (r & 1)] = Wm[i];
    }
    __syncthreads();

    int wave   = threadIdx.x >> 5;
    int lane   = threadIdx.x & 31;
    int tileM  = blockIdx.x * 8 + wave;
    if (tileM >= nTiles) return;

    int half   = lane >> 4;      // 0: K+0/K+1, 1: K+2/K+3
    int lanelo = lane & 15;      // M (for A) / N (for B)
    int row    = tileM * 16 + lanelo;
    int rowc   = row < N ? row : N - 1;           // clamp (N%16==0 normally)
    const float* hrow = h + (size_t)rowc * HDIM;

    v8f acc[8];
#pragma unroll
    for (int t = 0; t < 8; ++t)
#pragma unroll
        for (int v = 0; v < 8; ++v) acc[t][v] = 0.0f;

    for (int k = 0; k < HDIM; k += 4) {
        int ka = k + 2 * half;                    // this lane-half's K base
        v2f a = *(const v2f*)(hrow + ka);
        if (relu_in) { a.x = fmaxf(a.x, 0.0f); a.y = fmaxf(a.y, 0.0f); }
        const float* sWp = sW + (ka >> 1) * (2 * HDIM) + (lanelo << 1);
#pragma unroll
        for (int t = 0; t < 8; ++t) {
            v2f b = *(const v2f*)(sWp + t * 32);  // (W[ka][col], W[ka+1][col])
            acc[t] = __builtin_amdgcn_wmma_f32_16x16x4_f32(
                false, a, false, b, (short)0, acc[t], false, false);
        }
    }

    // C/D layout: VGPR v -> M = v + 8*half, N = lanelo
    float* zt0 = z + (size_t)(tileM * 16) * HDIM + lanelo;
    if (tileM * 16 + 16 <= N) {
#pragma unroll
        for (int t = 0; t < 8; ++t) {
            float* zt = zt0 + t * 16;
#pragma unroll
            for (int v = 0; v < 8; ++v)
                zt[(size_t)(v + 8 * half) * HDIM] = acc[t][v];
        }
    } else {
#pragma unroll
        for (int t = 0; t < 8; ++t) {
            float* zt = zt0 + t * 16;
#pragma unroll
            for (int v = 0; v < 8; ++v) {
                int m = v + 8 * half;
                if (tileM * 16 + m < N) zt[(size_t)m * HDIM] = acc[t][v];
            }
        }
    }
}

// out[n][f] = z[n][f]*selfn[n] + bc[f]   (self-loop term + bias, pre-atomics)
__global__ void k_agg_init(const float* __restrict__ z, const float* __restrict__ selfn,
                           const float* __restrict__ bc, float* __restrict__ out,
                           int total) {
    int idx = blockIdx.x * 256 + threadIdx.x;
    if (idx < total) {
        int n = idx >> 7;
        int f = idx & (HDIM - 1);
        out[idx] = z[idx] * selfn[n] + bc[f];
    }
}

// out[dst] += dinv[src]*dinv[dst] * z[src] ; one wave32 per edge, float4/lane
__global__ __launch_bounds__(256) void k_agg_edges(
        const i64* __restrict__ src, const i64* __restrict__ dst,
        const float* __restrict__ dinv, const float* __restrict__ z,
        float* __restrict__ out, int E) {
    int e = blockIdx.x * 8 + (threadIdx.x >> 5);
    if (e >= E) return;
    int lane = threadIdx.x & 31;
    int s = (int)src[e];
    int d = (int)dst[e];
    float w = dinv[s] * dinv[d];
    float4 v = ((const float4*)(z + (size_t)s * HDIM))[lane];
    float* op = out + (size_t)d * HDIM + lane * 4;
    atomicAdd(op + 0, w * v.x);
    atomicAdd(op + 1, w * v.y);
    atomicAdd(op + 2, w * v.z);
    atomicAdd(op + 3, w * v.w);
}

// pooled[batch[n]] += relu(h[n]) ; counts[batch[n]] += 1 ; one wave per node
__global__ __launch_bounds__(256) void k_pool(
        const float* __restrict__ h, const i64* __restrict__ batch,
        float* __restrict__ pooled, float* __restrict__ counts, int N) {
    int n = blockIdx.x * 8 + (threadIdx.x >> 5);
    if (n >= N) return;
    int lane = threadIdx.x & 31;
    int g = (int)batch[n];
    float4 v = ((const float4*)(h + (size_t)n * HDIM))[lane];
    float* pp = pooled + (size_t)g * HDIM + lane * 4;
    atomicAdd(pp + 0, fmaxf(v.x, 0.0f));
    atomicAdd(pp + 1, fmaxf(v.y, 0.0f));
    atomicAdd(pp + 2, fmaxf(v.z, 0.0f));
    atomicAdd(pp + 3, fmaxf(v.w, 0.0f));
    if (lane == 0) atomicAdd(&counts[g], 1.0f);
}

// readout MLP + log_softmax; one block (128 threads) per graph
__global__ __launch_bounds__(128) void k_readout(
        const float* __restrict__ pooled, const float* __restrict__ counts,
        const float* __restrict__ Wl1, const float* __restrict__ bl1,
        const float* __restrict__ Wl2, const float* __restrict__ bl2,
        float* __restrict__ out, int C) {
    __shared__ float sp[HDIM];
    __shared__ float sh[HDIM];
    __shared__ float sl[16];
    int g = blockIdx.x;
    int j = threadIdx.x;
    float inv = 1.0f / fmaxf(counts[g], 1.0f);
    sp[j] = pooled[(size_t)g * HDIM + j] * inv;
    __syncthreads();
    float acc = bl1[j];
    for (int k = 0; k < HDIM; ++k) acc += sp[k] * Wl1[k * HDIM + j];
    sh[j] = fmaxf(acc, 0.0f);
    __syncthreads();
    if (j < C) {
        float a2 = bl2[j];
        for (int k = 0; k < HDIM; ++k) a2 += sh[k] * Wl2[k * C + j];
        sl[j] = a2;
    }
    __syncthreads();
    if (j == 0) {
        float m = -INFINITY;
        for (int c = 0; c < C; ++c) m = fmaxf(m, sl[c]);
        float s = 0.0f;
        for (int c = 0; c < C; ++c) s += expf(sl[c] - m);
        float lse = m + logf(s);
        for (int c = 0; c < C; ++c) out[(size_t)g * C + c] = sl[c] - lse;
    }
}

// ---------------------------------------------------------------------------
extern "C" void kernel_launch(void* const* d_in, const int* in_sizes, int n_in,
                              void* d_out, int out_size, void* d_ws, size_t ws_size,
                              hipStream_t stream) {
    const float* x    = (const float*)d_in[0];
    const i64*   eidx = (const i64*)d_in[1];   // [2,E] int64
    const i64*   batch= (const i64*)d_in[2];   // [N]  int64
    const float* W1   = (const float*)d_in[3];
    const float* b1   = (const float*)d_in[4];
    const float* Wr1  = (const float*)d_in[5];
    const float* br1  = (const float*)d_in[6];
    const float* W2   = (const float*)d_in[7];
    const float* b2   = (const float*)d_in[8];
    const float* Wr   = (const float*)d_in[9];
    const float* br   = (const float*)d_in[10];
    const float* Wl1  = (const float*)d_in[11];
    const float* bl1  = (const float*)d_in[12];
    const float* Wl2  = (const float*)d_in[13];
    const float* bl2  = (const float*)d_in[14];

    const int N = in_sizes[2];
    const int E = in_sizes[1] / 2;
    const int C = in_sizes[14];
    const int G = out_size / C;
    const i64* esrc = eidx;
    const i64* edst = eidx + E;

    // workspace layout (floats)
    float* ws    = (float*)d_ws;
    size_t NH    = (size_t)N * HDIM;
    float* bufA  = ws;  ws += NH;
    float* bufB  = ws;  ws += NH;
    float* Wc    = ws;  ws += 3 * HDIM * HDIM;
    float* bc    = ws;  ws += 3 * HDIM;
    float* deg   = ws;  ws += N;
    float* dinv  = ws;  ws += N;
    float* selfn = ws;  ws += N;
    float* pooled= ws;  ws += (size_t)G * HDIM;
    float* counts= ws;  ws += G;

    hipMemsetAsync(deg,    0, (size_t)N * sizeof(float), stream);
    hipMemsetAsync(pooled, 0, (size_t)G * HDIM * sizeof(float), stream);
    hipMemsetAsync(counts, 0, (size_t)G * sizeof(float), stream);

    k_fold<<<(3 * HDIM * HDIM + 3 * HDIM + 255) / 256, 256, 0, stream>>>(
        W1, Wr1, W2, Wr, b1, br1, b2, br, Wc, bc);
    k_deg<<<(E + 255) / 256, 256, 0, stream>>>(edst, deg, E);
    k_norms<<<(N + 255) / 256, 256, 0, stream>>>(deg, dinv, selfn, N);

    const int nTiles     = (N + 15) / 16;
    const int gemmBlocks = (nTiles + 7) / 8;
    const int totNH      = N * HDIM;
    const int initBlocks = (totNH + 255) / 256;
    const int edgeBlocks = (E + 7) / 8;

    // layer 1: in = x (no relu on input)
    k_gemm_wmma<<<gemmBlocks, 256, 0, stream>>>(x, Wc, bufA, nTiles, N, 0);
    k_agg_init<<<initBlocks, 256, 0, stream>>>(bufA, selfn, bc, bufB, totNH);
    k_agg_edges<<<edgeBlocks, 256, 0, stream>>>(esrc, edst, dinv, bufA, bufB, E);

    // layer 2: relu folded into GEMM input load
    k_gemm_wmma<<<gemmBlocks, 256, 0, stream>>>(bufB, Wc + HDIM * HDIM, bufA, nTiles, N, 1);
    k_agg_init<<<initBlocks, 256, 0, stream>>>(bufA, selfn, bc + HDIM, bufB, totNH);
    k_agg_edges<<<edgeBlocks, 256, 0, stream>>>(esrc, edst, dinv, bufA, bufB, E);

    // layer 3
    k_gemm_wmma<<<gemmBlocks, 256, 0, stream>>>(bufB, Wc + 2 * HDIM * HDIM, bufA, nTiles, N, 1);
    k_agg_init<<<initBlocks, 256, 0, stream>>>(bufA, selfn, bc + 2 * HDIM, bufB, totNH);
    k_agg_edges<<<edgeBlocks, 256, 0, stream>>>(esrc, edst, dinv, bufA, bufB, E);

    // mean pool (relu folded into read) + readout MLP + log_softmax
    k_pool<<<(N + 7) / 8, 256, 0, stream>>>(bufB, batch, pooled, counts, N);
    k_readout<<<G, HDIM, 0, stream>>>(pooled, counts, Wl1, bl1, Wl2, bl2,
                                      (float*)d_out, C);
}